// CrossGraphMatchingModel_6871947674188
// MI455X (gfx1250) — compile-verified
//
#include <hip/hip_runtime.h>
#include <hip/hip_bf16.h>
#include <math.h>

typedef float v2f __attribute__((ext_vector_type(2)));
typedef float v8f __attribute__((ext_vector_type(8)));
typedef unsigned int u32x4 __attribute__((ext_vector_type(4)));
typedef int i32x8 __attribute__((ext_vector_type(8)));
typedef int i32x4 __attribute__((ext_vector_type(4)));

#define NQ 128
#define ND 1024

// ---------------------------------------------------------------- utilities
__global__ void fill_kernel(float* p, int n, float v) {
  int t = blockIdx.x * blockDim.x + threadIdx.x;
  if (t < n) p[t] = v;
}

__global__ void deg_count_kernel(const int* __restrict__ dst, float* deg, int E) {
  int t = blockIdx.x * blockDim.x + threadIdx.x;
  if (t < E) atomicAdd(&deg[dst[t]], 1.0f);
}

// ------------------------------------------------- fp32 WMMA GEMM (exact)
// C[M,N] = A[M,K] @ B[K,N] (+bias[n]) (opt relu).  M%16==0, N%64==0, K%64==0.
// One wave -> one 16x16 C tile via V_WMMA_F32_16X16X4_F32.
// A tile (16 x 64) is DMA'd global->LDS by the Tensor Data Mover with the
// descriptor pad feature inserting 4 DWORDs after every 64 DWORDs, which
// yields the bank-conflict-free row stride of 68 floats.
__global__ __launch_bounds__(128) void gemm_wmma(
    const float* __restrict__ A, int lda,
    const float* __restrict__ B, int ldb,
    float* __restrict__ C, int ldc, int K,
    const float* __restrict__ bias, int relu) {
  __shared__ float As[16 * 68];
  const int tid = threadIdx.x;
  const int wv = tid >> 5, lane = tid & 31;
  const int m0 = blockIdx.y * 16;
  const int n0 = blockIdx.x * 64 + wv * 16;
  const int lc = lane & 15, hi = lane >> 4, kk = hi * 2;

  // ---- invariant parts of the TDM descriptor (all wave-uniform -> SGPRs)
  const unsigned lds_base =
      (unsigned)(size_t)(void*)As;  // low 32 bits of generic ptr = DS address
  const unsigned long long gbase =
      (unsigned long long)(size_t)(A + (size_t)m0 * lda);
  i32x8 g1;
  // data_size=2 (4B) | pad_enable | pad_interval code 5 (64 DW) | pad_amount
  // code 3 (4 DW)
  g1[0] = (2 << 16) | (1 << 20) | (5 << 22) | (3 << 25);
  const unsigned td0 = 1u << 30, td1 = 1u << 30;  // huge: no OOB clipping
  g1[1] = (int)((td0 & 0xFFFFu) << 16);           // tensor_dim0[15:0]
  g1[2] = (int)((td0 >> 16) | ((td1 & 0xFFFFu) << 16));
  g1[3] = (int)((td1 >> 16) | (64u << 16));       // tile_dim0 = 64
  g1[4] = 16;                                     // tile_dim1 = 16, tile_dim2=0
  g1[5] = (int)(unsigned)lda;                     // tensor_dim0_stride (elems)
  g1[6] = 0;
  g1[7] = 0;
  const i32x4 gz4 = {0, 0, 0, 0};
  const i32x8 gz8 = {0, 0, 0, 0, 0, 0, 0, 0};

  v8f acc = {};
  for (int k0 = 0; k0 < K; k0 += 64) {
    __syncthreads();  // all waves done reading previous tile
    if (wv == 0) {
      unsigned long long ga = gbase + (unsigned long long)k0 * 4ull;
      u32x4 g0;
      g0[0] = 1u;                                  // one valid descriptor
      g0[1] = lds_base;                            // LDS dest byte address
      g0[2] = (unsigned)(ga & 0xFFFFFFFFull);      // global_addr[31:0]
      g0[3] = (unsigned)((ga >> 32) & 0x01FFFFFFull) | (2u << 30);  // type=2
      __builtin_amdgcn_tensor_load_to_lds(g0, g1, gz4, gz4, gz8, 0);
      __builtin_amdgcn_s_wait_tensorcnt(0);
    }
    __syncthreads();  // tile visible to all waves
    for (int kt = 0; kt < 64; kt += 4) {
      v2f a, b;
      a.x = As[lc * 68 + kt + kk];
      a.y = As[lc * 68 + kt + kk + 1];
      const float* Bp = B + (size_t)(k0 + kt + kk) * ldb + n0 + lc;
      b.x = Bp[0];
      b.y = Bp[ldb];
      acc = __builtin_amdgcn_wmma_f32_16x16x4_f32(false, a, false, b,
                                                  (short)0, acc, false, false);
    }
  }
  for (int r = 0; r < 8; ++r) {
    int gr = m0 + r + hi * 8;
    int gc = n0 + lc;
    float v = acc[r];
    if (bias) v += bias[gc];
    if (relu) v = fmaxf(v, 0.0f);
    C[(size_t)gr * ldc + gc] = v;
  }
}

// ---------------------------------------------------------------- GCN parts
__global__ void gcn_aggregate(const float* __restrict__ h,
                              const int* __restrict__ src,
                              const int* __restrict__ dst,
                              const float* __restrict__ deg,
                              float* agg, int E) {
  int t = blockIdx.x * blockDim.x + threadIdx.x;
  if (t >= E * 256) return;
  int e = t >> 8, d = t & 255;
  int s = src[e], dd = dst[e];
  float coef = rsqrtf(deg[s]) * rsqrtf(deg[dd]);
  atomicAdd(&agg[(size_t)dd * 256 + d], h[(size_t)s * 256 + d] * coef);
}

// outcat left half (ld 512): relu(agg + h/deg + b)
__global__ void gcn_finalize(const float* __restrict__ h,
                             const float* __restrict__ agg,
                             const float* __restrict__ deg,
                             const float* __restrict__ bias,
                             float* __restrict__ outcat, int N) {
  int t = blockIdx.x * blockDim.x + threadIdx.x;
  if (t >= N * 256) return;
  int n = t >> 8, d = t & 255;
  float v = agg[t] + h[t] / deg[n] + bias[d];
  outcat[(size_t)n * 512 + d] = fmaxf(v, 0.0f);
}

// ---------------------------------------------------------------- matching
__global__ void cos_kernel(const float* __restrict__ qcat,
                           const float* __restrict__ dcat,
                           const int* __restrict__ qi,
                           const int* __restrict__ di, int E,
                           float* __restrict__ cosv) {
  int wid = (blockIdx.x * blockDim.x + threadIdx.x) >> 5;
  int lane = threadIdx.x & 31;
  if (wid >= E) return;
  const float* qs = qcat + (size_t)qi[wid] * 512;
  const float* ds = dcat + (size_t)di[wid] * 512;
  float dot = 0.f, qn = 0.f, dn = 0.f;
  for (int k = lane; k < 256; k += 32) {
    float a = qs[k], b = ds[k];
    dot += a * b; qn += a * a; dn += b * b;
  }
  for (int m = 16; m; m >>= 1) {
    dot += __shfl_xor(dot, m, 32);
    qn  += __shfl_xor(qn, m, 32);
    dn  += __shfl_xor(dn, m, 32);
  }
  if (lane == 0) {
    float a = fmaxf(sqrtf(qn), 1e-8f);
    float b = fmaxf(sqrtf(dn), 1e-8f);
    cosv[wid] = dot / (a * b);
  }
}

__global__ void softmax_stats(const float* __restrict__ cosv, int E, float* red) {
  __shared__ float sm[256];
  int tid = threadIdx.x;
  float m = -1e30f;
  for (int k = tid; k < E; k += 256) m = fmaxf(m, cosv[k]);
  sm[tid] = m; __syncthreads();
  for (int s = 128; s; s >>= 1) {
    if (tid < s) sm[tid] = fmaxf(sm[tid], sm[tid + s]);
    __syncthreads();
  }
  float mx = sm[0]; __syncthreads();
  float sum = 0.f;
  for (int k = tid; k < E; k += 256) sum += __expf(cosv[k] - mx);
  sm[tid] = sum; __syncthreads();
  for (int s = 128; s; s >>= 1) {
    if (tid < s) sm[tid] += sm[tid + s];
    __syncthreads();
  }
  if (tid == 0) { red[0] = mx; red[1] = sm[0]; }
}

// scatter  w*ds into qcat right half, w*qs into dcat right half
__global__ void match_scatter(const float* __restrict__ cosv,
                              const float* __restrict__ red,
                              const int* __restrict__ qi,
                              const int* __restrict__ di,
                              float* qcat, float* dcat, int E) {
  int t = blockIdx.x * blockDim.x + threadIdx.x;
  if (t >= E * 256) return;
  int e = t >> 8, d = t & 255;
  float w = __expf(cosv[e] - red[0]) / red[1];
  int q = qi[e], dn = di[e];
  float qs = qcat[(size_t)q * 512 + d];
  float ds = dcat[(size_t)dn * 512 + d];
  atomicAdd(&qcat[(size_t)q * 512 + 256 + d], w * ds);
  atomicAdd(&dcat[(size_t)dn * 512 + 256 + d], w * qs);
}

// c2 = b1 @ w2 + b2   (256 outputs)
__global__ void c2_kernel(const float* __restrict__ b1,
                          const float* __restrict__ w2,
                          const float* __restrict__ b2, float* c2) {
  int n = threadIdx.x;
  float acc = b2[n];
  for (int k = 0; k < 512; ++k) acc += b1[k] * w2[k * 256 + n];
  c2[n] = acc;
}

// -------------------------------------------------- fused pair MLP (hot loop)
// pred[i,j] = relu( relu( relu(A2[i]+B2[j]) @ w3 + b3 ) @ w4 + b4 )
// block: 64 threads (2 waves); wave handles 16 j's; grid (ND/32, NQ).
__global__ __launch_bounds__(64) void pair_mlp(
    const float* __restrict__ A2, const float* __restrict__ B2,
    const float* __restrict__ w3, const float* __restrict__ b3,
    const float* __restrict__ w4, const float* __restrict__ b4,
    float* __restrict__ pred) {
  __shared__ float a2s[256];
  __shared__ float Us[2 * 16 * 260];  // padded rows -> conflict-free ds reads
  const int tid = threadIdx.x;
  const int wv = tid >> 5, lane = tid & 31;
  const int i = blockIdx.y;
  const int jbase = blockIdx.x * 32 + wv * 16;
  for (int d = tid; d < 256; d += 64) a2s[d] = A2[(size_t)i * 256 + d];
  __syncthreads();
  float* U = Us + wv * 16 * 260;
  for (int idx = lane; idx < 16 * 256; idx += 32) {
    int j = idx >> 8, d = idx & 255;
    float u = a2s[d] + B2[(size_t)(jbase + j) * 256 + d];
    U[j * 260 + d] = fmaxf(u, 0.0f);
  }
  __syncthreads();
  const int lc = lane & 15, hi = lane >> 4, kk = hi * 2;
  float psum[8];
  for (int r = 0; r < 8; ++r) psum[r] = 0.0f;
  for (int nt = 0; nt < 8; ++nt) {
    const int n0 = nt * 16;
    v8f acc = {};
    for (int k0 = 0; k0 < 256; k0 += 4) {
      v2f a, b;
      a.x = U[lc * 260 + k0 + kk];
      a.y = U[lc * 260 + k0 + kk + 1];
      b.x = w3[(size_t)(k0 + kk) * 128 + n0 + lc];
      b.y = w3[(size_t)(k0 + kk + 1) * 128 + n0 + lc];
      acc = __builtin_amdgcn_wmma_f32_16x16x4_f32(false, a, false, b,
                                                  (short)0, acc, false, false);
    }
    float w4v = w4[n0 + lc];
    float b3v = b3[n0 + lc];
    for (int r = 0; r < 8; ++r)
      psum[r] += fmaxf(acc[r] + b3v, 0.0f) * w4v;
  }
  float b4v = b4[0];
  for (int r = 0; r < 8; ++r) {
    float s = psum[r];
    s += __shfl_xor(s, 1, 32);
    s += __shfl_xor(s, 2, 32);
    s += __shfl_xor(s, 4, 32);
    s += __shfl_xor(s, 8, 32);
    if (lc == 0) {
      int jr = jbase + r + hi * 8;
      pred[(size_t)i * ND + jr] = fmaxf(s + b4v, 0.0f);
    }
  }
}

// ---------------------------------------------------------------- driver
extern "C" void kernel_launch(void* const* d_in, const int* in_sizes, int n_in,
                              void* d_out, int out_size, void* d_ws, size_t ws_size,
                              hipStream_t stream) {
  const float* qfeat = (const float*)d_in[0];
  const float* dfeat = (const float*)d_in[1];
  const int*   qedge = (const int*)d_in[2];
  const int*   dedge = (const int*)d_in[3];
  const int*   qi    = (const int*)d_in[4];
  const int*   di    = (const int*)d_in[5];
  const float* q_w0 = (const float*)d_in[6];
  const float* q_b0 = (const float*)d_in[7];
  const float* q_w1 = (const float*)d_in[8];
  const float* q_b1 = (const float*)d_in[9];
  const float* d_w0 = (const float*)d_in[10];
  const float* d_b0 = (const float*)d_in[11];
  const float* d_w1 = (const float*)d_in[12];
  const float* d_b1 = (const float*)d_in[13];
  const float* w1 = (const float*)d_in[14];
  const float* b1 = (const float*)d_in[15];
  const float* w2 = (const float*)d_in[16];
  const float* b2 = (const float*)d_in[17];
  const float* w3 = (const float*)d_in[18];
  const float* b3 = (const float*)d_in[19];
  const float* w4 = (const float*)d_in[20];
  const float* b4 = (const float*)d_in[21];

  const int Eq = in_sizes[2] / 2;
  const int Ed = in_sizes[3] / 2;
  const int Em = in_sizes[4];

  float* ws    = (float*)d_ws;
  float* h_q   = ws;                   // 32768
  float* h_d   = h_q + 32768;          // 262144
  float* deg_q = h_d + 262144;         // 128
  float* deg_d = deg_q + 128;          // 1024
  float* agg_q = deg_d + 1024;         // 32768
  float* agg_d = agg_q + 32768;        // 262144
  float* qcat1 = agg_d + 262144;       // 65536
  float* dcat1 = qcat1 + 65536;        // 524288
  float* cosv  = dcat1 + 524288;       // Em (8192)
  float* red   = cosv + 8192;          // 2 (+pad)
  float* W12a  = red + 64;             // 131072
  float* W12b  = W12a + 131072;        // 131072
  float* c2    = W12b + 131072;        // 256
  float* A2    = c2 + 256;             // 32768
  float* B2    = A2 + 32768;           // 262144

  float* pred   = (float*)d_out;
  float* qf_out = pred + (size_t)NQ * ND;       // [128,512]
  float* df_out = qf_out + (size_t)NQ * 512;    // [1024,512]

  // degrees (edge-count + self loop)
  fill_kernel<<<1, 128, 0, stream>>>(deg_q, NQ, 1.0f);
  fill_kernel<<<4, 256, 0, stream>>>(deg_d, ND, 1.0f);
  deg_count_kernel<<<(Eq + 255) / 256, 256, 0, stream>>>(qedge + Eq, deg_q, Eq);
  deg_count_kernel<<<(Ed + 255) / 256, 256, 0, stream>>>(dedge + Ed, deg_d, Ed);

  // fused first two MLP layers: W12a = w1[:512]@w2, W12b = w1[512:]@w2, c2
  gemm_wmma<<<dim3(4, 32), 128, 0, stream>>>(w1, 512, w2, 256, W12a, 256, 512, nullptr, 0);
  gemm_wmma<<<dim3(4, 32), 128, 0, stream>>>(w1 + 512 * 512, 512, w2, 256, W12b, 256, 512, nullptr, 0);
  c2_kernel<<<1, 256, 0, stream>>>(b1, w2, b2, c2);

  // ------------------------------ layer 1
  gemm_wmma<<<dim3(4, NQ / 16), 128, 0, stream>>>(qfeat, 256, q_w0, 256, h_q, 256, 256, nullptr, 0);
  gemm_wmma<<<dim3(4, ND / 16), 128, 0, stream>>>(dfeat, 256, d_w0, 256, h_d, 256, 256, nullptr, 0);
  fill_kernel<<<(NQ * 512 + 255) / 256, 256, 0, stream>>>(qcat1, NQ * 512, 0.0f);
  fill_kernel<<<(ND * 512 + 255) / 256, 256, 0, stream>>>(dcat1, ND * 512, 0.0f);
  fill_kernel<<<(NQ * 256 + 255) / 256, 256, 0, stream>>>(agg_q, NQ * 256, 0.0f);
  fill_kernel<<<(ND * 256 + 255) / 256, 256, 0, stream>>>(agg_d, ND * 256, 0.0f);
  gcn_aggregate<<<Eq, 256, 0, stream>>>(h_q, qedge, qedge + Eq, deg_q, agg_q, Eq);
  gcn_aggregate<<<Ed, 256, 0, stream>>>(h_d, dedge, dedge + Ed, deg_d, agg_d, Ed);
  gcn_finalize<<<NQ, 256, 0, stream>>>(h_q, agg_q, deg_q, q_b0, qcat1, NQ);
  gcn_finalize<<<ND, 256, 0, stream>>>(h_d, agg_d, deg_d, d_b0, dcat1, ND);
  cos_kernel<<<(Em * 32 + 255) / 256, 256, 0, stream>>>(qcat1, dcat1, qi, di, Em, cosv);
  softmax_stats<<<1, 256, 0, stream>>>(cosv, Em, red);
  match_scatter<<<Em, 256, 0, stream>>>(cosv, red, qi, di, qcat1, dcat1, Em);

  // ------------------------------ layer 2 (writes straight into d_out qf/df)
  gemm_wmma<<<dim3(4, NQ / 16), 128, 0, stream>>>(qcat1, 512, q_w1, 256, h_q, 256, 512, nullptr, 0);
  gemm_wmma<<<dim3(4, ND / 16), 128, 0, stream>>>(dcat1, 512, d_w1, 256, h_d, 256, 512, nullptr, 0);
  fill_kernel<<<(NQ * 512 + 255) / 256, 256, 0, stream>>>(qf_out, NQ * 512, 0.0f);
  fill_kernel<<<(ND * 512 + 255) / 256, 256, 0, stream>>>(df_out, ND * 512, 0.0f);
  fill_kernel<<<(NQ * 256 + 255) / 256, 256, 0, stream>>>(agg_q, NQ * 256, 0.0f);
  fill_kernel<<<(ND * 256 + 255) / 256, 256, 0, stream>>>(agg_d, ND * 256, 0.0f);
  gcn_aggregate<<<Eq, 256, 0, stream>>>(h_q, qedge, qedge + Eq, deg_q, agg_q, Eq);
  gcn_aggregate<<<Ed, 256, 0, stream>>>(h_d, dedge, dedge + Ed, deg_d, agg_d, Ed);
  gcn_finalize<<<NQ, 256, 0, stream>>>(h_q, agg_q, deg_q, q_b1, qf_out, NQ);
  gcn_finalize<<<ND, 256, 0, stream>>>(h_d, agg_d, deg_d, d_b1, df_out, ND);
  cos_kernel<<<(Em * 32 + 255) / 256, 256, 0, stream>>>(qf_out, df_out, qi, di, Em, cosv);
  softmax_stats<<<1, 256, 0, stream>>>(cosv, Em, red);
  match_scatter<<<Em, 256, 0, stream>>>(cosv, red, qi, di, qf_out, df_out, Em);

  // ------------------------------ final: A2/B2 then fused pair MLP
  gemm_wmma<<<dim3(4, NQ / 16), 128, 0, stream>>>(qf_out, 512, W12a, 256, A2, 256, 512, c2, 0);
  gemm_wmma<<<dim3(4, ND / 16), 128, 0, stream>>>(df_out, 512, W12b, 256, B2, 256, 512, nullptr, 0);
  pair_mlp<<<dim3(ND / 32, NQ), 64, 0, stream>>>(A2, B2, w3, b3, w4, b4, pred);
}